// Mamba2Layer_41154376630419
// MI455X (gfx1250) — compile-verified
//
#include <hip/hip_runtime.h>
#include <hip/hip_bf16.h>

// ---------------------------------------------------------------------------
// Mamba2 layer forward for gfx1250 (MI455X), fp32 throughout, using
// V_WMMA_F32_16X16X4_F32 for every GEMM-shaped computation.
// LDS tiles are stored K-contiguous per fragment row (B/X/S transposed) so
// each WMMA operand is a single aligned ds_load_b64.
// ---------------------------------------------------------------------------

#define BATCH   4
#define LSEQ    4096
#define DMODEL  1024
#define DINNER  2048
#define NHEADS  16
#define HDIM    128
#define DSTATE  64
#define CHUNKT  256
#define NCHUNK  (LSEQ / CHUNKT)     // 16
#define DPROJ   4240                // 2*DINNER + CONVDIM_extra + NHEADS
#define CONVDIM 2176                // DINNER + 2*DSTATE
#define DTOFF   (DINNER + CONVDIM)  // 4224: dt columns in zxbcdt
#define XCB     2048                // B columns inside conv output
#define XCC     2112                // C columns inside conv output

#define LPAD    68                  // LDS stride (floats): even + stride%64==4
#define XPAD    260                 // 256-wide rows: even + stride%64==4

typedef float v2f __attribute__((ext_vector_type(2)));
typedef float v8f __attribute__((ext_vector_type(8)));

__device__ __forceinline__ v8f wmma_f32_4(v2f a, v2f b, v8f c) {
  // D = A(16x4,f32) * B(4x16,f32) + C(16x16,f32)
  return __builtin_amdgcn_wmma_f32_16x16x4_f32(false, a, false, b, (short)0, c,
                                               false, false);
}

__device__ __forceinline__ v8f zero8() {
  v8f z = {0.f, 0.f, 0.f, 0.f, 0.f, 0.f, 0.f, 0.f};
  return z;
}

// ---------------------------------------------------------------------------
// Generic fp32 WMMA GEMM: C[M,N] = A[M,K] * B[K,N].  M%64==0, K%64==0,
// N guarded.  64x64 block, 8 waves; wave w -> rows 16*(w&3), cols 32*(w>>2).
// A staged as As[m][k]; B staged transposed as BsT[n][k] so both WMMA
// operands read two consecutive, 8B-aligned floats (ds_load_b64).
// ---------------------------------------------------------------------------
__launch_bounds__(256)
__global__ void gemm_f32_wmma(const float* __restrict__ A,
                              const float* __restrict__ B,
                              float* __restrict__ C, int M, int N, int K) {
  __shared__ float As[64][LPAD];
  __shared__ float BsT[64][LPAD];   // [n][k]
  const int tid  = threadIdx.x;
  const int lane = tid & 31;
  const int wave = tid >> 5;
  const int row0 = blockIdx.y * 64;
  const int col0 = blockIdx.x * 64;
  const int mt = wave & 3;          // 16-row tile within block
  const int nb = (wave >> 2) * 32;  // 32-col strip within block
  const int lm = lane & 15;
  const int kh = (lane >> 4) << 1;  // 0 or 2 (K half select)
  const int lr = tid >> 2;          // staging row 0..63
  const int lc = (tid & 3) << 4;    // staging col 0/16/32/48

  v8f acc0 = zero8();
  v8f acc1 = zero8();

  for (int k0 = 0; k0 < K; k0 += 64) {
    const float* ga = A + (size_t)(row0 + lr) * K + (k0 + lc);
#pragma unroll
    for (int v = 0; v < 4; ++v) {
      float4 t4 = *(const float4*)(ga + v * 4);
      As[lr][lc + v * 4 + 0] = t4.x;
      As[lr][lc + v * 4 + 1] = t4.y;
      As[lr][lc + v * 4 + 2] = t4.z;
      As[lr][lc + v * 4 + 3] = t4.w;
    }
    if (col0 + 64 <= N) {
      const float* gb = B + (size_t)(k0 + lr) * N + (col0 + lc);
#pragma unroll
      for (int v = 0; v < 4; ++v) {
        float4 t4 = *(const float4*)(gb + v * 4);
        BsT[lc + v * 4 + 0][lr] = t4.x;   // transpose into [n][k]
        BsT[lc + v * 4 + 1][lr] = t4.y;
        BsT[lc + v * 4 + 2][lr] = t4.z;
        BsT[lc + v * 4 + 3][lr] = t4.w;
      }
    } else {
#pragma unroll
      for (int v = 0; v < 16; ++v) {
        const int n = lc + v;
        float val = 0.f;
        if (col0 + n < N) val = B[(size_t)(k0 + lr) * N + (col0 + n)];
        BsT[n][lr] = val;
      }
    }
    // prefetch next K tile while this one is consumed
    if (k0 + 64 < K) {
      __builtin_prefetch(A + (size_t)(row0 + lr) * K + (k0 + 64 + lc), 0, 1);
      __builtin_prefetch(B + (size_t)(k0 + 64 + lr) * N + col0 + (tid & 3) * 16, 0, 1);
    }
    __syncthreads();
#pragma unroll
    for (int kk = 0; kk < 64; kk += 4) {
      v2f a;
      a.x = As[16 * mt + lm][kk + kh];
      a.y = As[16 * mt + lm][kk + kh + 1];
      v2f b0;
      b0.x = BsT[nb + lm][kk + kh];
      b0.y = BsT[nb + lm][kk + kh + 1];
      v2f b1;
      b1.x = BsT[nb + 16 + lm][kk + kh];
      b1.y = BsT[nb + 16 + lm][kk + kh + 1];
      acc0 = wmma_f32_4(a, b0, acc0);
      acc1 = wmma_f32_4(a, b1, acc1);
    }
    __syncthreads();
  }
  const int grow = row0 + 16 * mt + ((lane >> 4) << 3);
  const int gc0  = col0 + nb + lm;
#pragma unroll
  for (int r = 0; r < 8; ++r) {
    if (gc0 < N)      C[(size_t)(grow + r) * N + gc0]      = acc0[r];
    if (gc0 + 16 < N) C[(size_t)(grow + r) * N + gc0 + 16] = acc1[r];
  }
}

// ---------------------------------------------------------------------------
// dt softplus + per-chunk inclusive cumsum of dA = -exp(A_log)*dt
// grid (NCHUNK, NHEADS, BATCH) x 256
// ---------------------------------------------------------------------------
__launch_bounds__(256)
__global__ void dt_cumsum_kernel(const float* __restrict__ zx,
                                 const float* __restrict__ dt_bias,
                                 const float* __restrict__ A_log,
                                 float* __restrict__ dts,
                                 float* __restrict__ acum) {
  __shared__ float s[CHUNKT];
  const int c = blockIdx.x, h = blockIdx.y, b = blockIdx.z;
  const int t = threadIdx.x;
  const size_t row = (size_t)b * LSEQ + (size_t)c * CHUNKT + t;
  const float x  = zx[row * DPROJ + DTOFF + h] + dt_bias[h];
  const float sp = (x > 20.f) ? x : log1pf(expf(x));
  dts[row * NHEADS + h] = sp;
  s[t] = -expf(A_log[h]) * sp;
  __syncthreads();
  for (int o = 1; o < CHUNKT; o <<= 1) {
    const float v = (t >= o) ? s[t - o] : 0.f;
    __syncthreads();
    s[t] += v;
    __syncthreads();
  }
  acum[((((size_t)b * NHEADS + h) * NCHUNK + c) << 8) + t] = s[t];
}

// ---------------------------------------------------------------------------
// Causal depthwise conv (taps=4, left pad 3) + SiLU, elementwise.
// ---------------------------------------------------------------------------
__launch_bounds__(256)
__global__ void conv_silu_kernel(const float* __restrict__ zx,
                                 const float* __restrict__ cw,
                                 const float* __restrict__ cb,
                                 float* __restrict__ xcv) {
  const long long idx   = (long long)blockIdx.x * 256 + threadIdx.x;
  const long long total = (long long)BATCH * LSEQ * CONVDIM;
  if (idx >= total) return;
  const int ch = (int)(idx % CONVDIM);
  const long long bt = idx / CONVDIM;
  const int t = (int)(bt % LSEQ);
  const long long brow = bt - t;  // b*LSEQ
  float acc = cb[ch];
#pragma unroll
  for (int k = 0; k < 4; ++k) {
    const int tsrc = t - 3 + k;
    if (tsrc >= 0)
      acc += cw[k * CONVDIM + ch] * zx[(size_t)(brow + tsrc) * DPROJ + DINNER + ch];
  }
  xcv[idx] = acc / (1.f + expf(-acc));  // SiLU
}

// ---------------------------------------------------------------------------
// SSD chunk kernel: per (chunk, head, batch) workgroup, 8 waves.
//   phase 1: Y_diag = (C B^T * L) @ Xd  (waves pair row-tiles i and 15-i)
//   phase 2: chunk end-state S[n,p] = sum_t B[t,n]*exp(ac_last-ac[t])*Xd[t,p]
// x*dt kept in LDS transposed as XdT[p][t] so the WMMA B-operand is a single
// aligned ds_load_b64; B/C fragments stream from global (L2-resident).
// ---------------------------------------------------------------------------
__launch_bounds__(256)
__global__ void ssd_chunk_kernel(const float* __restrict__ xc,
                                 const float* __restrict__ dts,
                                 const float* __restrict__ acumg,
                                 float* __restrict__ y,
                                 float* __restrict__ states) {
  __shared__ float XdT[HDIM][XPAD];    // ~133 KB, dt-scaled x, [p][t]
  __shared__ float ac[CHUNKT];
  __shared__ float dec[CHUNKT];
  __shared__ float Gs[8][16][18];      // per-wave G' staging, [t][s]
  const int c = blockIdx.x, h = blockIdx.y, b = blockIdx.z;
  const int tid  = threadIdx.x;
  const int lane = tid & 31;
  const int wave = tid >> 5;
  const int lm = lane & 15;
  const int kh = (lane >> 4) << 1;
  const size_t row0 = (size_t)b * LSEQ + (size_t)c * CHUNKT;

  const float* acp = acumg + ((((size_t)b * NHEADS + h) * NCHUNK + c) << 8);
  const float alast = acp[CHUNKT - 1];
  {
    const float a = acp[tid];
    ac[tid]  = a;
    dec[tid] = expf(alast - a);
  }
  for (int i = tid; i < CHUNKT * HDIM; i += 256) {
    const int t = i >> 7, p = i & 127;
    const float xv = xc[(row0 + t) * CONVDIM + h * HDIM + p];
    XdT[p][t] = xv * dts[(row0 + t) * NHEADS + h];
  }
  __syncthreads();

  // ---- phase 1: Y_diag ----
  for (int pass = 0; pass < 2; ++pass) {
    const int i = pass ? (15 - wave) : wave;
    const size_t crowA = (row0 + i * 16 + lm) * CONVDIM + XCC;  // C row base
    v8f Yacc[8];
#pragma unroll
    for (int q = 0; q < 8; ++q) Yacc[q] = zero8();

    for (int j = 0; j <= i; ++j) {
      const size_t browB = (row0 + j * 16 + lm) * CONVDIM + XCB;  // B row base
      v8f g = zero8();
#pragma unroll
      for (int kk = 0; kk < DSTATE; kk += 4) {
        const float2 ca  = *(const float2*)(xc + crowA + kk + kh);
        const float2 bb2 = *(const float2*)(xc + browB + kk + kh);
        v2f a;  a.x = ca.x;   a.y = ca.y;
        v2f bb; bb.x = bb2.x; bb.y = bb2.y;
        g = wmma_f32_4(a, bb, g);   // G[t,s] += C[t,n]*B[s,n]
      }
      // decay mask: L[t,s] = exp(ac[t]-ac[s]) for s<=t else 0
      const int trow = i * 16 + ((lane >> 4) << 3);
      const int scol = j * 16 + lm;
      const float as = ac[scol];
#pragma unroll
      for (int r = 0; r < 8; ++r) {
        const int t = trow + r;
        g[r] *= (scol <= t) ? expf(ac[t] - as) : 0.f;
      }
      // stage G' (D-layout) and read back as A-fragment
#pragma unroll
      for (int r = 0; r < 8; ++r)
        Gs[wave][((lane >> 4) << 3) + r][lm] = g[r];
      asm volatile("s_wait_dscnt 0" ::: "memory");
#pragma unroll
      for (int kk = 0; kk < 16; kk += 4) {
        v2f a;
        a.x = Gs[wave][lm][kk + kh];
        a.y = Gs[wave][lm][kk + kh + 1];
        const int s0 = j * 16 + kk + kh;
#pragma unroll
        for (int nt = 0; nt < 8; ++nt) {
          v2f bb;
          bb.x = XdT[nt * 16 + lm][s0];
          bb.y = XdT[nt * 16 + lm][s0 + 1];
          Yacc[nt] = wmma_f32_4(a, bb, Yacc[nt]);
        }
      }
      asm volatile("s_wait_dscnt 0" ::: "memory");
    }
    // store Y_diag into y (b, l, h*128+p)
    const size_t yrow = row0 + i * 16 + ((lane >> 4) << 3);
#pragma unroll
    for (int nt = 0; nt < 8; ++nt) {
      const size_t col = (size_t)h * HDIM + nt * 16 + lm;
#pragma unroll
      for (int r = 0; r < 8; ++r)
        y[(yrow + r) * DINNER + col] = Yacc[nt][r];
    }
  }

  // ---- phase 2: chunk end state ----
  {
    const int mtn = wave & 3;         // state-dim tile (16 rows of n)
    const int pb  = (wave >> 2) << 2; // p tiles pb..pb+3
    const int nst = mtn * 16 + lm;
    v8f S[4];
#pragma unroll
    for (int q = 0; q < 4; ++q) S[q] = zero8();
#pragma unroll 8
    for (int kk = 0; kk < CHUNKT; kk += 4) {
      const int t0 = kk + kh;
      v2f a;  // A[m=n][k=t] = B[t,n]*dec[t]
      a.x = xc[(row0 + t0)     * CONVDIM + XCB + nst] * dec[t0];
      a.y = xc[(row0 + t0 + 1) * CONVDIM + XCB + nst] * dec[t0 + 1];
#pragma unroll
      for (int q = 0; q < 4; ++q) {
        v2f bb;
        bb.x = XdT[(pb + q) * 16 + lm][t0];
        bb.y = XdT[(pb + q) * 16 + lm][t0 + 1];
        S[q] = wmma_f32_4(a, bb, S[q]);
      }
    }
    float* sp = states + ((((size_t)b * NCHUNK + c) * NHEADS + h) << 13);
    const int nr = mtn * 16 + ((lane >> 4) << 3);
#pragma unroll
    for (int q = 0; q < 4; ++q) {
      const int p = (pb + q) * 16 + lm;
#pragma unroll
      for (int r = 0; r < 8; ++r)
        sp[p * DSTATE + nr + r] = S[q][r];   // layout [p][n]
    }
  }
}

// ---------------------------------------------------------------------------
// Inter-chunk state recurrence (in-place exclusive scan over 16 chunks):
//   init[c] = S;  S = exp(A_last[c])*S + state[c]
// grid (NHEADS, BATCH) x 256; each thread owns 32 of the 8192 state elems.
// ---------------------------------------------------------------------------
__launch_bounds__(256)
__global__ void state_scan_kernel(const float* __restrict__ acumg,
                                  float* __restrict__ states) {
  const int h = blockIdx.x, b = blockIdx.y;
  const int tid = threadIdx.x;
  float S[32];
#pragma unroll
  for (int q = 0; q < 32; ++q) S[q] = 0.f;
  for (int c = 0; c < NCHUNK; ++c) {
    const float dc =
        expf(acumg[((((size_t)b * NHEADS + h) * NCHUNK + c) << 8) + CHUNKT - 1]);
    float* stc = states + ((((size_t)b * NCHUNK + c) * NHEADS + h) << 13);
#pragma unroll
    for (int q = 0; q < 32; ++q) {
      const int e   = tid + 256 * q;
      const float v = stc[e];
      stc[e] = S[q];              // exclusive prefix -> initial state of chunk
      S[q]   = dc * S[q] + v;
    }
  }
}

// ---------------------------------------------------------------------------
// Y_off = (C @ S_init) * exp(acum[t]);  y += Y_off + D*x_dt
// S_init staged transposed as SsT[p][n] -> contiguous WMMA B-operand.
// grid (NCHUNK, NHEADS, BATCH) x 256
// ---------------------------------------------------------------------------
__launch_bounds__(256)
__global__ void yoff_kernel(const float* __restrict__ xc,
                            const float* __restrict__ dts,
                            const float* __restrict__ acumg,
                            const float* __restrict__ states,
                            const float* __restrict__ Dvec,
                            float* __restrict__ y) {
  __shared__ float SsT[HDIM][LPAD];   // [p][n], ~35 KB
  __shared__ float ac[CHUNKT];
  const int c = blockIdx.x, h = blockIdx.y, b = blockIdx.z;
  const int tid  = threadIdx.x;
  const int lane = tid & 31;
  const int wave = tid >> 5;
  const int lm = lane & 15;
  const int kh = (lane >> 4) << 1;
  const size_t row0 = (size_t)b * LSEQ + (size_t)c * CHUNKT;
  const float* acp = acumg + ((((size_t)b * NHEADS + h) * NCHUNK + c) << 8);
  ac[tid] = acp[tid];
  const float* sp = states + ((((size_t)b * NCHUNK + c) * NHEADS + h) << 13);
  for (int i = tid; i < DSTATE * HDIM; i += 256) {
    const int p = i >> 6, n = i & 63;
    SsT[p][n] = sp[p * DSTATE + n];   // global is [p][n] already: coalesced
  }
  __syncthreads();
  const float Dh = Dvec[h];

  for (int pass = 0; pass < 2; ++pass) {
    const int i = wave + 8 * pass;
    const size_t crow = (row0 + i * 16 + lm) * CONVDIM + XCC;
    v8f acc[8];
#pragma unroll
    for (int q = 0; q < 8; ++q) acc[q] = zero8();
#pragma unroll
    for (int kk = 0; kk < DSTATE; kk += 4) {
      const float2 ca = *(const float2*)(xc + crow + kk + kh);
      v2f a; a.x = ca.x; a.y = ca.y;
#pragma unroll
      for (int nt = 0; nt < 8; ++nt) {
        v2f bb;
        bb.x = SsT[nt * 16 + lm][kk + kh];
        bb.y = SsT[nt * 16 + lm][kk + kh + 1];
        acc[nt] = wmma_f32_4(a, bb, acc[nt]);
      }
    }
    const int tl0 = i * 16 + ((lane >> 4) << 3);
#pragma unroll
    for (int r = 0; r < 8; ++r) {
      const int tl = tl0 + r;
      const size_t grow = row0 + tl;
      const float sdo = expf(ac[tl]);
      const float dtv = dts[grow * NHEADS + h];
#pragma unroll
      for (int nt = 0; nt < 8; ++nt) {
        const int p = nt * 16 + lm;
        const float xdt = xc[grow * CONVDIM + h * HDIM + p] * dtv;
        const size_t yi = grow * DINNER + h * HDIM + p;
        y[yi] += acc[nt][r] * sdo + Dh * xdt;
      }
    }
  }
}

// ---------------------------------------------------------------------------
// y = (y*z) * rsqrt(mean((y*z)^2) + eps) * norm_scale, per token row.
// ---------------------------------------------------------------------------
__launch_bounds__(256)
__global__ void gate_norm_kernel(const float* __restrict__ zx,
                                 const float* __restrict__ nscale,
                                 float* __restrict__ y) {
  __shared__ float sred[256];
  const size_t row = blockIdx.x;
  const float* zrow = zx + row * DPROJ;  // z = first DINNER columns
  float* yrow = y + row * DINNER;
  float g[8];
  float ss = 0.f;
#pragma unroll
  for (int q = 0; q < 8; ++q) {
    const int col = threadIdx.x + 256 * q;
    const float v = yrow[col] * zrow[col];
    g[q] = v;
    ss += v * v;
  }
  sred[threadIdx.x] = ss;
  __syncthreads();
  for (int o = 128; o > 0; o >>= 1) {
    if (threadIdx.x < o) sred[threadIdx.x] += sred[threadIdx.x + o];
    __syncthreads();
  }
  const float rms = rsqrtf(sred[0] * (1.f / DINNER) + 1e-5f);
#pragma unroll
  for (int q = 0; q < 8; ++q) {
    const int col = threadIdx.x + 256 * q;
    yrow[col] = g[q] * rms * nscale[col];
  }
}

// ---------------------------------------------------------------------------
extern "C" void kernel_launch(void* const* d_in, const int* in_sizes, int n_in,
                              void* d_out, int out_size, void* d_ws,
                              size_t ws_size, hipStream_t stream) {
  const float* u       = (const float*)d_in[0];
  const float* W_in    = (const float*)d_in[1];
  const float* conv_w  = (const float*)d_in[2];
  const float* conv_b  = (const float*)d_in[3];
  const float* dt_bias = (const float*)d_in[4];
  const float* A_log   = (const float*)d_in[5];
  const float* Dvec    = (const float*)d_in[6];
  const float* nscale  = (const float*)d_in[7];
  const float* W_out   = (const float*)d_in[8];
  float* out = (float*)d_out;

  // workspace layout (floats)
  float* zx     = (float*)d_ws;                              // 16384*4240
  float* xcv    = zx     + (size_t)16384 * DPROJ;            // 16384*2176
  float* dts    = xcv    + (size_t)16384 * CONVDIM;          // 16384*16
  float* acum   = dts    + (size_t)16384 * NHEADS;           // 262144
  float* states = acum   + (size_t)BATCH * NHEADS * NCHUNK * CHUNKT;  // 8388608
  float* yb     = states + (size_t)BATCH * NCHUNK * NHEADS * HDIM * DSTATE; // 16384*2048

  const dim3 blk(256);
  const int M = BATCH * LSEQ;  // 16384

  // 1) in-projection GEMM: zx = u @ W_in
  gemm_f32_wmma<<<dim3((DPROJ + 63) / 64, M / 64), blk, 0, stream>>>(
      u, W_in, zx, M, DPROJ, DMODEL);
  // 2) dt softplus + per-chunk cumsum of A*dt
  dt_cumsum_kernel<<<dim3(NCHUNK, NHEADS, BATCH), blk, 0, stream>>>(
      zx, dt_bias, A_log, dts, acum);
  // 3) depthwise conv + SiLU
  conv_silu_kernel<<<dim3((unsigned)(((long long)M * CONVDIM + 255) / 256)),
                     blk, 0, stream>>>(zx, conv_w, conv_b, xcv);
  // 4) SSD intra-chunk (Y_diag + chunk states)
  ssd_chunk_kernel<<<dim3(NCHUNK, NHEADS, BATCH), blk, 0, stream>>>(
      xcv, dts, acum, yb, states);
  // 5) inter-chunk state recurrence
  state_scan_kernel<<<dim3(NHEADS, BATCH), blk, 0, stream>>>(acum, states);
  // 6) Y_off + D*x contribution
  yoff_kernel<<<dim3(NCHUNK, NHEADS, BATCH), blk, 0, stream>>>(
      xcv, dts, acum, states, Dvec, yb);
  // 7) gate by z + RMS norm
  gate_norm_kernel<<<dim3(M), blk, 0, stream>>>(zx, nscale, yb);
  // 8) out-projection GEMM: out = yb @ W_out
  gemm_f32_wmma<<<dim3(DMODEL / 64, M / 64), blk, 0, stream>>>(
      yb, W_out, out, M, DMODEL, DINNER);
}